// GNN_16999480557861
// MI455X (gfx1250) — compile-verified
//
#include <hip/hip_runtime.h>
#include <hip/hip_bf16.h>

typedef __attribute__((ext_vector_type(2))) float v2f;
typedef __attribute__((ext_vector_type(8))) float v8f;

#define GNN_N 100000
#define GNN_E 1600000
#define GNN_D 128
#define GNN_DP (GNN_D + 4)   // LDS row padding to dodge 64-bank conflicts

// ---------------------------------------------------------------------------
// Zero-fill (workspace is poisoned 0xAA; must zero deterministically each call)
// ---------------------------------------------------------------------------
__global__ void gnn_fill_zero(float* __restrict__ p, int n) {
    int i = blockIdx.x * blockDim.x + threadIdx.x;
    if (i < n) p[i] = 0.0f;
}

// ---------------------------------------------------------------------------
// Degree count: cnt[dst[e]] += 1
// ---------------------------------------------------------------------------
__global__ void gnn_count(const int* __restrict__ dst, float* __restrict__ cnt, int E) {
    int e = blockIdx.x * blockDim.x + threadIdx.x;
    if (e < E) atomicAdd(&cnt[dst[e]], 1.0f);
}

// in-place: cnt[i] = 1 / max(cnt[i], 1)
__global__ void gnn_recip(float* __restrict__ cnt, int n) {
    int i = blockIdx.x * blockDim.x + threadIdx.x;
    if (i < n) cnt[i] = 1.0f / fmaxf(cnt[i], 1.0f);
}

// ---------------------------------------------------------------------------
// Scatter-add: one wave (32 lanes) per edge, each lane moves 4 consecutive
// floats of the 128-float source row. Coalesced 512B gather per edge.
// ---------------------------------------------------------------------------
__global__ __launch_bounds__(256)
void gnn_scatter(const float* __restrict__ h, const int* __restrict__ src,
                 const int* __restrict__ dst, float* __restrict__ agg, int E) {
    long long gid = (long long)blockIdx.x * blockDim.x + threadIdx.x;
    int e    = (int)(gid >> 5);
    int lane = (int)(gid & 31);
    if (e >= E) return;
    int s = src[e];
    int d = dst[e];
    const float4 v = ((const float4*)(h + (long long)s * GNN_D))[lane];
    float* ap = agg + (long long)d * GNN_D + lane * 4;
    atomicAdd(ap + 0, v.x);
    atomicAdd(ap + 1, v.y);
    atomicAdd(ap + 2, v.z);
    atomicAdd(ap + 3, v.w);
}

// ---------------------------------------------------------------------------
// Fused SAGE layer GEMM:
//   hout = relu_opt( (agg * inv_cnt) @ Wl^T + hin @ Wr^T + bias )
// Block = 256 threads = 8 waves; block owns 16 nodes, wave w owns features
// [16w, 16w+16). 64 x v_wmma_f32_16x16x4_f32 per wave (32 K-steps x 2 inputs).
//
// f32 16x16x4 fragment layouts (ISA 7.12.2):
//   A (16x4): lane L -> m = L%16, kpair = 2*(L/16); a = {A[m][k+kp], A[m][k+kp+1]}
//   B (4x16): lane L -> n = L%16, kpair = 2*(L/16); b = {B[k+kp][n], B[k+kp+1][n]}
//   C/D (16x16): lane L -> n = L%16; vgpr r -> m = r + 8*(L/16)
// With B[k][j] = W[j*128 + k] this computes in @ W^T.
// ---------------------------------------------------------------------------
__global__ __launch_bounds__(256)
void gnn_sage_gemm(const float* __restrict__ agg, const float* __restrict__ hin,
                   const float* __restrict__ inv_cnt,
                   const float* __restrict__ Wl, const float* __restrict__ Wr,
                   const float* __restrict__ bias, float* __restrict__ hout,
                   int relu) {
    __shared__ float s_mean[16][GNN_DP];
    __shared__ float s_h[16][GNN_DP];

    const int block_node = blockIdx.x * 16;

    // Stage 16 rows of mean-aggregate and input into LDS (float4 moves).
    for (int i = threadIdx.x; i < 16 * (GNN_D / 4); i += 256) {
        int m = i / (GNN_D / 4);
        int c = i % (GNN_D / 4);
        long long node = block_node + m;
        float4 a  = ((const float4*)(agg + node * GNN_D))[c];
        float  ic = inv_cnt[node];
        a.x *= ic; a.y *= ic; a.z *= ic; a.w *= ic;
        float4 h4 = ((const float4*)(hin + node * GNN_D))[c];
        *(float4*)&s_mean[m][c * 4] = a;
        *(float4*)&s_h[m][c * 4]    = h4;
    }
    __syncthreads();

    const int wave = threadIdx.x >> 5;   // 0..7
    const int lane = threadIdx.x & 31;
    const int half = lane >> 4;          // 0 or 1 -> K-pair select
    const int lm   = lane & 15;          // m for A, n for B/C/D
    const int jcol = wave * 16 + lm;     // output feature owned by this lane

    v8f acc = {};

    #pragma unroll
    for (int k = 0; k < GNN_D; k += 4) {
        const int kb = k + half * 2;
        v2f a_m, b_l, a_h, b_r;
        a_m[0] = s_mean[lm][kb];
        a_m[1] = s_mean[lm][kb + 1];
        b_l[0] = Wl[(long long)jcol * GNN_D + kb];
        b_l[1] = Wl[(long long)jcol * GNN_D + kb + 1];
        acc = __builtin_amdgcn_wmma_f32_16x16x4_f32(
            false, a_m, false, b_l, (short)0, acc, false, false);

        a_h[0] = s_h[lm][kb];
        a_h[1] = s_h[lm][kb + 1];
        b_r[0] = Wr[(long long)jcol * GNN_D + kb];
        b_r[1] = Wr[(long long)jcol * GNN_D + kb + 1];
        acc = __builtin_amdgcn_wmma_f32_16x16x4_f32(
            false, a_h, false, b_r, (short)0, acc, false, false);
    }

    const float bj = bias[jcol];
    #pragma unroll
    for (int r = 0; r < 8; ++r) {
        int m = r + half * 8;
        float v = acc[r] + bj;
        if (relu) v = fmaxf(v, 0.0f);
        hout[(long long)(block_node + m) * GNN_D + jcol] = v;
    }
}

// ---------------------------------------------------------------------------
// Orchestration
// ---------------------------------------------------------------------------
extern "C" void kernel_launch(void* const* d_in, const int* in_sizes, int n_in,
                              void* d_out, int out_size, void* d_ws, size_t ws_size,
                              hipStream_t stream) {
    const float* x   = (const float*)d_in[0];
    const int*   ei  = (const int*)d_in[1];        // [2, E] int32
    const float* Wl1 = (const float*)d_in[2];
    const float* Wr1 = (const float*)d_in[3];
    const float* b1  = (const float*)d_in[4];
    const float* Wl2 = (const float*)d_in[5];
    const float* Wr2 = (const float*)d_in[6];
    const float* b2  = (const float*)d_in[7];
    const float* Wl3 = (const float*)d_in[8];
    const float* Wr3 = (const float*)d_in[9];
    const float* b3  = (const float*)d_in[10];

    const int* src = ei;
    const int* dst = ei + GNN_E;

    // Workspace layout (floats): cnt[N] | agg[N*D] | h1[N*D] | h2[N*D]
    float* cnt = (float*)d_ws;
    float* agg = cnt + GNN_N;
    float* h1  = agg + (long long)GNN_N * GNN_D;
    float* h2  = h1  + (long long)GNN_N * GNN_D;
    float* h3  = (float*)d_out;

    const int ND = GNN_N * GNN_D;
    const dim3 blk(256);
    const int  g_fill_nd  = (ND + 255) / 256;
    const int  g_fill_n   = (GNN_N + 255) / 256;
    const int  g_edges    = (GNN_E + 255) / 256;
    const int  g_scatter  = (GNN_E * 32) / 256;   // wave per edge
    const int  g_gemm     = GNN_N / 16;           // 6250 tiles

    // Degree + reciprocal (shared by all 3 layers)
    gnn_fill_zero<<<g_fill_n, blk, 0, stream>>>(cnt, GNN_N);
    gnn_count<<<g_edges, blk, 0, stream>>>(dst, cnt, GNN_E);
    gnn_recip<<<g_fill_n, blk, 0, stream>>>(cnt, GNN_N);

    // Layer 1: x -> h1 (relu)
    gnn_fill_zero<<<g_fill_nd, blk, 0, stream>>>(agg, ND);
    gnn_scatter<<<g_scatter, blk, 0, stream>>>(x, src, dst, agg, GNN_E);
    gnn_sage_gemm<<<g_gemm, blk, 0, stream>>>(agg, x, cnt, Wl1, Wr1, b1, h1, 1);

    // Layer 2: h1 -> h2 (relu)
    gnn_fill_zero<<<g_fill_nd, blk, 0, stream>>>(agg, ND);
    gnn_scatter<<<g_scatter, blk, 0, stream>>>(h1, src, dst, agg, GNN_E);
    gnn_sage_gemm<<<g_gemm, blk, 0, stream>>>(agg, h1, cnt, Wl2, Wr2, b2, h2, 1);

    // Layer 3: h2 -> out (no relu)
    gnn_fill_zero<<<g_fill_nd, blk, 0, stream>>>(agg, ND);
    gnn_scatter<<<g_scatter, blk, 0, stream>>>(h2, src, dst, agg, GNN_E);
    gnn_sage_gemm<<<g_gemm, blk, 0, stream>>>(agg, h2, cnt, Wl3, Wr3, b3, h3, 0);
}